// MuHopf_5540507811919
// MI455X (gfx1250) — compile-verified
//
#include <hip/hip_runtime.h>

// ---------------------------------------------------------------------------
// MuHopf oscillator bank: zs[step][b][d] = r_n * exp(i*phi_n), 256 steps.
// Store-bandwidth-bound (268 MB output @ 23.3 TB/s ~= 11.6 us floor).
// Strategy: per-thread register-resident recurrence, rotation trick replaces
// per-step sin/cos with 4 FMAs, 128-bit non-temporal stores (write-once
// stream exceeds the 192 MB L2). No matrix structure exists in this op, so
// the VMEM/VALU mix (not WMMA) is the roofline-correct path.
// ---------------------------------------------------------------------------

typedef __attribute__((ext_vector_type(4))) float v4f;

constexpr int   kB     = 128;
constexpr int   kD     = 1024;
constexpr int   kSteps = 256;
constexpr float kDT    = 0.01f;
constexpr float kTwoPi = 6.28318530717958647692f;

constexpr int VEC      = 4;                 // elements (columns) per thread
constexpr int kThreads = kB * kD / VEC;     // 32768 threads = 1024 wave32
constexpr int kBlock   = 128;               // 4 waves per workgroup

__global__ __launch_bounds__(kBlock)
void muhopf_stream(const float* __restrict__ xg,     // [B,D]
                   const float* __restrict__ rg,     // [B,D]
                   const float* __restrict__ pg,     // [B,D]
                   const float* __restrict__ wg,     // [D] omegas_raw
                   float* __restrict__ zs,           // [steps,B,D] cplx
                   float* __restrict__ rf,           // [B,D]
                   float* __restrict__ pf)           // [B,D]
{
    const int gid = blockIdx.x * kBlock + threadIdx.x;   // [0, B*D/4)
    const int e   = gid * VEC;                            // first flat element
    const int d   = e & (kD - 1);                         // column of elem 0

    // 128-bit coalesced input loads
    const v4f xv = *(const v4f*)(xg + e);
    const v4f rv = *(const v4f*)(rg + e);
    const v4f pv = *(const v4f*)(pg + e);
    const v4f wv = *(const v4f*)(wg + d);

    float r[VEC], mu[VEC], dphi[VEC];
    float c[VEC], s[VEC], cd[VEC], sd[VEC];

    #pragma unroll
    for (int i = 0; i < VEC; ++i) {
        r[i]  = rv[i];
        mu[i] = 5.0f * (1.0f + xv[i]);                    // 5*(MU0 + x)
        // omega = (sigmoid(w) * (5.0-0.5) + 0.5) * 2*pi ; dphi = omega*dt
        // fast rcp instead of IEEE divide: init-only, 1-ulp is irrelevant here
        const float sig   = __builtin_amdgcn_rcpf(1.0f + __expf(-wv[i]));
        const float omega = (sig * 4.5f + 0.5f) * kTwoPi;
        dphi[i] = omega * kDT;
        // rotation constants (once), and phase state at first emitted step
        __sincosf(dphi[i],          &sd[i], &cd[i]);
        __sincosf(pv[i] + dphi[i],  &s[i],  &c[i]);
    }

    float*       out        = zs + (size_t)e * 2;         // complex interleave
    const size_t stepStride = (size_t)kB * kD * 2;        // floats per step

    #pragma unroll 4
    for (int n = 0; n < kSteps; ++n) {
        v4f z0, z1;
        #pragma unroll
        for (int i = 0; i < VEC; ++i) {
            // r_{n+1} = r + (mu - r^2) * r * dt
            float rr = r[i];
            rr   = fmaf((mu[i] - rr * rr) * rr, kDT, rr);
            r[i] = rr;

            const float zr = rr * c[i];
            const float zi = rr * s[i];

            // phase rotation for next step: (c,s) *= (cd, sd)
            const float cn = fmaf(c[i], cd[i], -(s[i] * sd[i]));
            const float sn = fmaf(s[i], cd[i],   c[i] * sd[i]);
            c[i] = cn; s[i] = sn;

            if (i < 2) { z0[2 * i]       = zr; z0[2 * i + 1]       = zi; }
            else       { z1[2 * (i - 2)] = zr; z1[2 * (i - 2) + 1] = zi; }
        }
        // 2 x global_store_b128, non-temporal (bypass/limit L2 residency)
        __builtin_nontemporal_store(z0, (v4f*)(out));
        __builtin_nontemporal_store(z1, (v4f*)(out + 4));
        out += stepStride;
    }

    // final state outputs: phi_f = phi0 + NUM_STEPS*dphi (phi not needed in loop)
    v4f rvo, pvo;
    #pragma unroll
    for (int i = 0; i < VEC; ++i) {
        rvo[i] = r[i];
        pvo[i] = fmaf((float)kSteps, dphi[i], pv[i]);
    }
    __builtin_nontemporal_store(rvo, (v4f*)(rf + e));
    __builtin_nontemporal_store(pvo, (v4f*)(pf + e));
}

extern "C" void kernel_launch(void* const* d_in, const int* in_sizes, int n_in,
                              void* d_out, int out_size, void* d_ws, size_t ws_size,
                              hipStream_t stream) {
    const float* x   = (const float*)d_in[0];   // [B,D]
    const float* r   = (const float*)d_in[1];   // [B,D]
    const float* phi = (const float*)d_in[2];   // [B,D]
    const float* w   = (const float*)d_in[3];   // [D]

    float* out = (float*)d_out;
    float* zs  = out;                                          // 256*B*D*2 f32
    float* rf  = out + (size_t)kSteps * kB * kD * 2;           // B*D f32
    float* pf  = rf + (size_t)kB * kD;                         // B*D f32

    muhopf_stream<<<dim3(kThreads / kBlock), dim3(kBlock), 0, stream>>>(
        x, r, phi, w, zs, rf, pf);
}